// RNNTLoss_82721070121387
// MI455X (gfx1250) — compile-verified
//
#include <hip/hip_runtime.h>
#include <stdint.h>

// Problem constants (fixed by the reference's setup_inputs)
#define BB   8
#define TT   512
#define UU   50
#define UP1  51
#define VV   1024
// Diagonal-major compact layout
#define WPAD 52                   // padded row width (u index 0..50, +1 pad) -> 208B rows
#define CHD  32                   // diagonals per async chunk
#define NCHK 18                   // chunks; covers d = 0..575 (need 0..561)
#define DROWS (CHD * NCHK)        // 576 rows
#define CV4  (CHD * WPAD / 4)     // 416 float4 per chunk per array (13 * 32 exactly)

__device__ __forceinline__ float lae(float a, float b) {
    float m = fmaxf(a, b);
    return m + log1pf(expf(-fabsf(a - b)));
}

// ---------------------------------------------------------------------------
// Kernel 1: gather blank/emit from the huge logits tensor into compact
// anti-diagonal-major arrays in workspace.
//   BL[b][d][u] = blank[b][d-u-1][u]   (operand of cell (t=d-u, u))
//   EM[b][d][u] = emit [b][d-u][u-1]   (operand of cell (t=d-u, u))
// ---------------------------------------------------------------------------
__global__ void rnnt_gather(const float* __restrict__ x,
                            const int*   __restrict__ y,
                            float* __restrict__ EM,
                            float* __restrict__ BL) {
    const int bt  = blockIdx.x;          // b*TT + t
    const int b   = bt / TT;
    const int t   = bt - b * TT;
    const int tid = threadIdx.x;
    const float* xrow = x + (size_t)bt * UP1 * VV;
    const size_t base = (size_t)b * DROWS * WPAD;

    if (tid < UP1) {
        // blank[b][t][tid] is consumed by cell (t+1, tid) -> diagonal d = t+1+tid
        float v = xrow[(size_t)tid * VV];
        BL[base + (size_t)(t + 1 + tid) * WPAD + tid] = v;
    } else if (tid >= 64 && tid < 64 + UU) {
        int j   = tid - 64;                        // 0..49
        int lab = y[b * UU + j];                   // labels in [1, V)
        float v = xrow[(size_t)j * VV + (size_t)lab];
        v = 1.002f * v - 0.002f * v;               // FastEmit fwd (value-identical)
        // emit[b][t][j] is consumed by cell (t, j+1) -> diagonal d = t+j+1
        EM[base + (size_t)(t + j + 1) * WPAD + (j + 1)] = v;
    }
}

// ---------------------------------------------------------------------------
// Kernel 2: anti-diagonal wavefront DP, one wave32 per batch.
// Operands streamed via gfx1250 async global->LDS copies, double buffered.
// ---------------------------------------------------------------------------
__global__ void __launch_bounds__(32)
rnnt_dp(const float* __restrict__ EM,
        const float* __restrict__ BL,
        const float* __restrict__ x,
        const int*   __restrict__ xl,
        const int*   __restrict__ yl,
        float* __restrict__ LOSS) {
    __shared__ float4 em_lds[2][CV4];   // 2 x 6656 B
    __shared__ float4 bl_lds[2][CV4];   // total 26.6 KB LDS

    const int b = blockIdx.x;
    const int l = threadIdx.x;          // lane 0..31 (wave32)

    const float4* emsrc = (const float4*)(EM + (size_t)b * DROWS * WPAD);
    const float4* blsrc = (const float4*)(BL + (size_t)b * DROWS * WPAD);

    // Async-copy one 32-diagonal chunk (416 float4 per array) into LDS buffer.
    // 13 uniform iterations x 2 arrays = 26 async instructions per chunk;
    // two chunks in flight = 52 <= 63 (ASYNCcnt is 6 bits).
    auto issue_chunk = [&](int c, int buf) {
        const int base4 = c * CV4;
        for (int i = l; i < CV4; i += 32) {
            unsigned long long ga =
                (unsigned long long)(uintptr_t)(emsrc + base4 + i);
            unsigned la =
                (unsigned)(unsigned long long)(uintptr_t)(&em_lds[buf][i]);
            asm volatile("global_load_async_to_lds_b128 %0, %1, off"
                         :: "v"(la), "v"(ga) : "memory");
            ga = (unsigned long long)(uintptr_t)(blsrc + base4 + i);
            la = (unsigned)(unsigned long long)(uintptr_t)(&bl_lds[buf][i]);
            asm volatile("global_load_async_to_lds_b128 %0, %1, off"
                         :: "v"(la), "v"(ga) : "memory");
        }
    };

    int myxl = xl[b]; myxl = myxl < 1 ? 1 : (myxl > TT ? TT : myxl);
    int myyl = yl[b]; myyl = myyl < 0 ? 0 : (myyl > UU ? UU : myyl);
    const int tcap = myxl - 1;
    const int u0 = l;
    const int u1 = l + 32;
    const int u1c = (u1 <= UU) ? u1 : UU;   // clamped LDS read index

    float A0 = 0.0f;   // rolling diagonal alpha for u0
    float A1 = 0.0f;   // rolling diagonal alpha for u1
    float sv = 0.0f;   // captured alpha[tcap][yl]

    issue_chunk(0, 0);
    issue_chunk(1, 1);
    asm volatile("s_wait_asynccnt 0x1A" ::: "memory");  // chunk 0 resident

    for (int c = 0; c < NCHK; ++c) {
        const float* emf = (const float*)(&em_lds[c & 1][0]);
        const float* blf = (const float*)(&bl_lds[c & 1][0]);
        #pragma unroll 4
        for (int k = 0; k < CHD; ++k) {
            const int d = c * CHD + k;
            // read previous-diagonal neighbors BEFORE any update
            float s_up0 = __shfl_up(A0, 1, 32);
            float s_up1 = __shfl_up(A1, 1, 32);
            float s_hi  = __shfl(A0, 31, 32);
            float pA0m1 = s_up0;                       // alpha[t][u0-1]
            float pA1m1 = (l == 0) ? s_hi : s_up1;     // alpha[t][u1-1]

            float em0 = emf[k * WPAD + u0];
            float bl0 = blf[k * WPAD + u0];
            float em1 = emf[k * WPAD + u1c];
            float bl1 = blf[k * WPAD + u1c];

            int t0 = d - u0;
            if (t0 >= 0 && t0 < TT) {
                float v;
                if (u0 == 0)      v = (t0 == 0) ? 0.0f : (A0 + bl0);
                else if (t0 == 0) v = pA0m1 + em0;                     // cumsum row
                else              v = lae(A0 + bl0, pA0m1 + em0);
                A0 = v;
                if (t0 == tcap && u0 == myyl) sv = v;
            }
            int t1 = d - u1;
            if (u1 <= UU && t1 >= 0 && t1 < TT) {
                float v = (t1 == 0) ? (pA1m1 + em1)
                                    : lae(A1 + bl1, pA1m1 + em1);
                A1 = v;
                if (t1 == tcap && u1 == myyl) sv = v;
            }
        }
        if (c + 1 < NCHK) {
            // my ds reads of this buffer must retire before async rewrites it
            asm volatile("s_wait_dscnt 0x0" ::: "memory");
            if (c + 2 < NCHK) {
                issue_chunk(c + 2, c & 1);
                asm volatile("s_wait_asynccnt 0x1A" ::: "memory"); // chunk c+1 ready
            } else {
                asm volatile("s_wait_asynccnt 0x0" ::: "memory");  // drain
            }
        }
    }

    float af = __shfl(sv, myyl & 31, 32);   // lane owning u == yl
    if (l == 0) {
        size_t bi = (((size_t)b * TT + (size_t)tcap) * UP1 + (size_t)myyl) * VV;
        LOSS[b] = -(af + x[bi]);            // final_blank read straight from x
    }
}

// ---------------------------------------------------------------------------
// Kernel 3: deterministic fixed-order mean of per-batch losses.
// ---------------------------------------------------------------------------
__global__ void rnnt_reduce(const float* __restrict__ LOSS,
                            float* __restrict__ out) {
    if (threadIdx.x == 0 && blockIdx.x == 0) {
        float s = 0.0f;
        for (int i = 0; i < BB; ++i) s += LOSS[i];
        out[0] = s * (1.0f / BB);
    }
}

extern "C" void kernel_launch(void* const* d_in, const int* in_sizes, int n_in,
                              void* d_out, int out_size, void* d_ws, size_t ws_size,
                              hipStream_t stream) {
    (void)in_sizes; (void)n_in; (void)out_size; (void)ws_size;
    const float* x  = (const float*)d_in[0];
    const int*   y  = (const int*)d_in[1];
    const int*   xl = (const int*)d_in[2];
    const int*   yl = (const int*)d_in[3];

    float* EM   = (float*)d_ws;                         // [BB][DROWS][WPAD]
    float* BL   = EM + (size_t)BB * DROWS * WPAD;       // [BB][DROWS][WPAD]
    float* LOSS = BL + (size_t)BB * DROWS * WPAD;       // [BB]

    rnnt_gather<<<BB * TT, 128, 0, stream>>>(x, y, EM, BL);
    rnnt_dp<<<BB, 32, 0, stream>>>(EM, BL, x, xl, yl, LOSS);
    rnnt_reduce<<<1, 32, 0, stream>>>(LOSS, (float*)d_out);
}